// GNN_29583734735123
// MI455X (gfx1250) — compile-verified
//
#include <hip/hip_runtime.h>
#include <hip/hip_bf16.h>
#include <math.h>

// ---------------- problem constants (match reference) ----------------
#define FEATLEN 64
#define DIMA    256
#define BATCH   2048
#define ATOMS   30
#define NNODE   (BATCH * ATOMS)     // 61440
#define NEDGE   (4 * NNODE)         // 245760
#define SSIDE   2
#define KNEI    32

typedef __attribute__((ext_vector_type(2))) float v2f;
typedef __attribute__((ext_vector_type(8))) float v8f;

// ---------------- small helper kernels ----------------

__global__ void gnn_deg_kernel(const int* __restrict__ esrc, const int* __restrict__ edst,
                               float* __restrict__ degout, float* __restrict__ degin) {
    int e = blockIdx.x * blockDim.x + threadIdx.x;
    if (e >= NEDGE) return;
    atomicAdd(&degout[esrc[e]], 1.0f);
    atomicAdd(&degin[edst[e]], 1.0f);
}

__global__ void gnn_norm_kernel(const float* __restrict__ degout, const float* __restrict__ degin,
                                float* __restrict__ nsrc, float* __restrict__ ndst) {
    int i = blockIdx.x * blockDim.x + threadIdx.x;
    if (i >= NNODE) return;
    nsrc[i] = rsqrtf(fmaxf(degout[i], 1.0f));
    ndst[i] = rsqrtf(fmaxf(degin[i], 1.0f));
}

// h0[i][j] = number of the 4 feature slots equal to j  (== one_hot(...).sum(axis=1))
__global__ void gnn_onehot_kernel(const int* __restrict__ nf, float* __restrict__ h0) {
    int i = blockIdx.x * blockDim.x + threadIdx.x;
    if (i >= NNODE) return;
    int f0 = nf[i * 4 + 0], f1 = nf[i * 4 + 1], f2 = nf[i * 4 + 2], f3 = nf[i * 4 + 3];
    float* row = h0 + (size_t)i * FEATLEN;
    for (int j = 0; j < FEATLEN; ++j)
        row[j] = (float)((f0 == j) + (f1 == j) + (f2 == j) + (f3 == j));
}

// agg[dst][d] += h[src][d] * nsrc[src]   (one block per edge, blockDim == D)
__global__ void gnn_scatter_kernel(const float* __restrict__ h, const float* __restrict__ nsrc,
                                   const int* __restrict__ esrc, const int* __restrict__ edst,
                                   float* __restrict__ agg, int D) {
    int e = blockIdx.x;
    int d = threadIdx.x;
    int s = esrc[e], t = edst[e];
    atomicAdd(&agg[(size_t)t * D + d], h[(size_t)s * D + d] * nsrc[s]);
}

// ---------------- generic WMMA f32 GEMM: C[M,256] = act( diag(scale) * A_gathered[M,K] @ W[K,256] + bias ) ----------------
// blockDim = 128 (4 waves). Each block: 16 rows; wave w: columns [64w, 64w+64).
__launch_bounds__(128)
__global__ void gnn_gemm256_wmma(const float* __restrict__ A, const int* __restrict__ rowidx,
                                 const float* __restrict__ rowscale,
                                 const float* __restrict__ W, const float* __restrict__ bias,
                                 float* __restrict__ C, int K, int do_relu) {
    const int wave = threadIdx.x >> 5;
    const int lane = threadIdx.x & 31;
    const int mloc = lane & 15;
    const int kb   = (lane >> 4) << 1;   // 0 for lanes 0-15, 2 for lanes 16-31
    const int nloc = lane & 15;
    const int nbase = wave * 64;

    const int row  = blockIdx.x * 16 + mloc;
    const int arow = rowidx ? rowidx[row] : row;
    const float scale = rowscale ? rowscale[row] : 1.0f;
    const float* __restrict__ arp = A + (size_t)arow * K;

    v8f acc0 = {}, acc1 = {}, acc2 = {}, acc3 = {};
    for (int k = 0; k < K; k += 4) {
        v2f a;
        a.x = arp[k + kb] * scale;
        a.y = arp[k + kb + 1] * scale;
        const float* __restrict__ w0 = W + (size_t)(k + kb) * DIMA + nbase + nloc;
        const float* __restrict__ w1 = W + (size_t)(k + kb + 1) * DIMA + nbase + nloc;
        v2f b0; b0.x = w0[0];  b0.y = w1[0];
        v2f b1; b1.x = w0[16]; b1.y = w1[16];
        v2f b2; b2.x = w0[32]; b2.y = w1[32];
        v2f b3; b3.x = w0[48]; b3.y = w1[48];
        acc0 = __builtin_amdgcn_wmma_f32_16x16x4_f32(false, a, false, b0, (short)0, acc0, false, false);
        acc1 = __builtin_amdgcn_wmma_f32_16x16x4_f32(false, a, false, b1, (short)0, acc1, false, false);
        acc2 = __builtin_amdgcn_wmma_f32_16x16x4_f32(false, a, false, b2, (short)0, acc2, false, false);
        acc3 = __builtin_amdgcn_wmma_f32_16x16x4_f32(false, a, false, b3, (short)0, acc3, false, false);
    }

    // C/D layout: VGPR r -> row (tile + r) for lanes 0-15, (tile + r + 8) for lanes 16-31, col = lane&15
    const int rh = (lane >> 4) << 3;
    const int rowbase = blockIdx.x * 16 + rh;
    const float bz = 0.0f;
    float bv0 = bias ? bias[nbase + nloc]      : bz;
    float bv1 = bias ? bias[nbase + 16 + nloc] : bz;
    float bv2 = bias ? bias[nbase + 32 + nloc] : bz;
    float bv3 = bias ? bias[nbase + 48 + nloc] : bz;
#pragma unroll
    for (int r = 0; r < 8; ++r) {
        float* crow = C + (size_t)(rowbase + r) * DIMA + nbase + nloc;
        float v0 = acc0[r] + bv0;
        float v1 = acc1[r] + bv1;
        float v2 = acc2[r] + bv2;
        float v3 = acc3[r] + bv3;
        if (do_relu) {
            v0 = fmaxf(v0, 0.0f); v1 = fmaxf(v1, 0.0f);
            v2 = fmaxf(v2, 0.0f); v3 = fmaxf(v3, 0.0f);
        }
        crow[0] = v0; crow[16] = v1; crow[32] = v2; crow[48] = v3;
    }
}

// sum of per-row L2 norms of h2 (one block per row)
__global__ void gnn_rownorm_kernel(const float* __restrict__ h2, float* __restrict__ sumnorm) {
    __shared__ float red[DIMA];
    int row = blockIdx.x, d = threadIdx.x;
    float v = h2[(size_t)row * DIMA + d];
    red[d] = v * v;
    __syncthreads();
    for (int off = DIMA / 2; off > 0; off >>= 1) {
        if (d < off) red[d] += red[d + off];
        __syncthreads();
    }
    if (d == 0) atomicAdd(sumnorm, sqrtf(red[0]));
}

// sum-pool 30 atoms per graph, scale by factor = sqrt(256) * N / sum_of_norms
__global__ void gnn_pool_kernel(const float* __restrict__ h2, const float* __restrict__ sumnorm,
                                float* __restrict__ out_full, float* __restrict__ out_atom) {
    int b = blockIdx.x, d = threadIdx.x;
    float s = 0.0f;
    for (int a = 0; a < ATOMS; ++a)
        s += h2[(size_t)(b * ATOMS + a) * DIMA + d];
    float factor = 16.0f * (float)NNODE / sumnorm[0];
    float v = s * factor;
    out_full[(size_t)b * 512 + d] = v;                 // mol_embedding[:, :256]
    out_atom[(size_t)b * DIMA + d] = v;                // mol_atom_embedding
}

// target_embedding = mean_s(src) + mean_s(tgt)
__global__ void gnn_target_kernel(const float* __restrict__ src_emb, const float* __restrict__ tgt_emb,
                                  float* __restrict__ te) {
    int t = blockIdx.x * blockDim.x + threadIdx.x;
    if (t >= BATCH * DIMA) return;
    int b = t >> 8, d = t & 255;
    float v = 0.5f * (src_emb[(size_t)b * DIMA + d] + src_emb[(size_t)(BATCH + b) * DIMA + d]
                    + tgt_emb[(size_t)b * DIMA + d] + tgt_emb[(size_t)(BATCH + b) * DIMA + d]);
    te[t] = v;
}

// nei_idx[(s*B+b)*K + k] = neighbor_matrix[id_list[s*B+b]][k]
__global__ void gnn_neiidx_kernel(const int* __restrict__ ids, const int* __restrict__ nmat,
                                  int* __restrict__ nidx) {
    int t = blockIdx.x * blockDim.x + threadIdx.x;
    if (t >= SSIDE * BATCH * KNEI) return;
    int sb = t >> 5, k = t & 31;
    nidx[t] = nmat[(size_t)ids[sb] * KNEI + k];
}

// attention over 32 neighbors + mean over s; writes mre and mol_embedding[:,256:]
__global__ void gnn_attention_kernel(const float* __restrict__ src_emb, const float* __restrict__ nei_emb,
                                     const float* __restrict__ te,
                                     const float* __restrict__ att_w, const float* __restrict__ att_b,
                                     float* __restrict__ mre, float* __restrict__ out_full) {
    __shared__ float red[DIMA];
    __shared__ float logit[KNEI];
    int b = blockIdx.x, d = threadIdx.x;
    float aw = att_w[d];
    float ab = att_b[0];
    float targ = te[(size_t)b * DIMA + d];
    float mres = 0.0f;
    for (int s = 0; s < SSIDE; ++s) {
        const float* item = src_emb + (size_t)(s * BATCH + b) * DIMA;
        float it = item[d];
        const float* nbase = nei_emb + (size_t)(s * BATCH + b) * KNEI * DIMA;
        for (int k = 0; k < KNEI; ++k) {
            float p = it * nbase[(size_t)k * DIMA + d] * targ * aw;
            red[d] = p;
            __syncthreads();
            for (int off = DIMA / 2; off > 0; off >>= 1) {
                if (d < off) red[d] += red[d + off];
                __syncthreads();
            }
            if (d == 0) logit[k] = red[0] + ab;
            __syncthreads();
        }
        if (d == 0) {
            float mx = logit[0];
            for (int k = 1; k < KNEI; ++k) mx = fmaxf(mx, logit[k]);
            float ssum = 0.0f;
            for (int k = 0; k < KNEI; ++k) { float e = expf(logit[k] - mx); logit[k] = e; ssum += e; }
            float inv = 1.0f / ssum;
            for (int k = 0; k < KNEI; ++k) logit[k] *= inv;
        }
        __syncthreads();
        float accd = 0.0f;
        for (int k = 0; k < KNEI; ++k) accd += logit[k] * nbase[(size_t)k * DIMA + d];
        mres += 0.5f * (it + accd);
        __syncthreads();
    }
    mre[(size_t)b * DIMA + d] = mres;
    out_full[(size_t)b * 512 + 256 + d] = mres;        // mol_embedding[:, 256:]
}

// ---------------- host-side orchestration ----------------

extern "C" void kernel_launch(void* const* d_in, const int* in_sizes, int n_in,
                              void* d_out, int out_size, void* d_ws, size_t ws_size,
                              hipStream_t stream) {
    (void)in_sizes; (void)n_in; (void)out_size; (void)ws_size;
    const int*   node_feat = (const int*)d_in[0];
    const int*   edge_src  = (const int*)d_in[1];
    const int*   edge_dst  = (const int*)d_in[2];
    /* graph_ids d_in[3] unused: contiguous repeat */
    const int*   id_list   = (const int*)d_in[4];
    const int*   id_tgt    = (const int*)d_in[5];
    const float* all_mfea  = (const float*)d_in[6];
    const int*   nmat      = (const int*)d_in[7];
    const float* W0        = (const float*)d_in[8];
    const float* b0        = (const float*)d_in[9];
    const float* W1        = (const float*)d_in[10];
    const float* b1        = (const float*)d_in[11];
    const float* fea_emb   = (const float*)d_in[12];
    const float* att_w     = (const float*)d_in[13];
    const float* att_b     = (const float*)d_in[14];
    const float* ccl_w     = (const float*)d_in[15];
    const float* ccl_b     = (const float*)d_in[16];
    float* out = (float*)d_out;

    // workspace carve-out (floats)
    float* w        = (float*)d_ws;
    float* deg_out  = w;
    float* deg_in   = deg_out + NNODE;
    float* nsrc     = deg_in  + NNODE;
    float* ndst     = nsrc    + NNODE;
    float* h0       = ndst    + NNODE;
    float* agg1     = h0   + (size_t)NNODE * FEATLEN;
    float* h1       = agg1 + (size_t)NNODE * FEATLEN;
    float* agg2     = h1   + (size_t)NNODE * DIMA;
    float* h2       = agg2 + (size_t)NNODE * DIMA;
    float* sumnorm  = h2   + (size_t)NNODE * DIMA;
    float* src_emb  = sumnorm + 4;
    float* tgt_emb  = src_emb + (size_t)SSIDE * BATCH * DIMA;
    float* te       = tgt_emb + (size_t)SSIDE * BATCH * DIMA;
    float* mre      = te  + (size_t)BATCH * DIMA;
    float* nei_emb  = mre + (size_t)BATCH * DIMA;
    int*   nei_idx  = (int*)(nei_emb + (size_t)SSIDE * BATCH * KNEI * DIMA);

    // zero the accumulated buffers
    hipMemsetAsync(deg_out, 0, (size_t)2 * NNODE * sizeof(float), stream);
    hipMemsetAsync(agg1,    0, (size_t)NNODE * FEATLEN * sizeof(float), stream);
    hipMemsetAsync(agg2,    0, (size_t)NNODE * DIMA * sizeof(float), stream);
    hipMemsetAsync(sumnorm, 0, 4 * sizeof(float), stream);

    // ---- atom-level GCN ----
    gnn_deg_kernel<<<(NEDGE + 255) / 256, 256, 0, stream>>>(edge_src, edge_dst, deg_out, deg_in);
    gnn_norm_kernel<<<(NNODE + 255) / 256, 256, 0, stream>>>(deg_out, deg_in, nsrc, ndst);
    gnn_onehot_kernel<<<(NNODE + 255) / 256, 256, 0, stream>>>(node_feat, h0);

    gnn_scatter_kernel<<<NEDGE, FEATLEN, 0, stream>>>(h0, nsrc, edge_src, edge_dst, agg1, FEATLEN);
    gnn_gemm256_wmma<<<NNODE / 16, 128, 0, stream>>>(agg1, nullptr, ndst, W0, b0, h1, FEATLEN, 1);

    gnn_scatter_kernel<<<NEDGE, DIMA, 0, stream>>>(h1, nsrc, edge_src, edge_dst, agg2, DIMA);
    gnn_gemm256_wmma<<<NNODE / 16, 128, 0, stream>>>(agg2, nullptr, ndst, W1, b1, h2, DIMA, 0);

    gnn_rownorm_kernel<<<NNODE, DIMA, 0, stream>>>(h2, sumnorm);
    gnn_pool_kernel<<<BATCH, DIMA, 0, stream>>>(h2, sumnorm, out, out + (size_t)BATCH * 512);

    // ---- reaction-level KG aggregation ----
    gnn_gemm256_wmma<<<(SSIDE * BATCH) / 16, 128, 0, stream>>>(all_mfea, id_list, nullptr,
                                                               fea_emb, nullptr, src_emb, 40, 0);
    gnn_gemm256_wmma<<<(SSIDE * BATCH) / 16, 128, 0, stream>>>(all_mfea, id_tgt, nullptr,
                                                               fea_emb, nullptr, tgt_emb, 40, 0);
    gnn_target_kernel<<<(BATCH * DIMA + 255) / 256, 256, 0, stream>>>(src_emb, tgt_emb, te);

    gnn_neiidx_kernel<<<(SSIDE * BATCH * KNEI + 255) / 256, 256, 0, stream>>>(id_list, nmat, nei_idx);
    gnn_gemm256_wmma<<<(SSIDE * BATCH * KNEI) / 16, 128, 0, stream>>>(all_mfea, nei_idx, nullptr,
                                                                      fea_emb, nullptr, nei_emb, 40, 0);

    gnn_attention_kernel<<<BATCH, DIMA, 0, stream>>>(src_emb, nei_emb, te, att_w, att_b, mre, out);

    // mol_x_embedding = mre @ crosscl_w + crosscl_b   -> third output segment
    gnn_gemm256_wmma<<<BATCH / 16, 128, 0, stream>>>(mre, nullptr, nullptr, ccl_w, ccl_b,
                                                     out + (size_t)BATCH * 768, DIMA, 0);
}